// SegmentationAdapter_22703197127180
// MI455X (gfx1250) — compile-verified
//
#include <hip/hip_runtime.h>
#include <cstdint>

// ---------------------------------------------------------------------------
// Connected-component labeling + boxes for 16 x (512x512) masks.
// Phase 1: per-32x32-tile CCL in LDS; tile staged by ONE Tensor-Data-Mover
//          DMA (tensor_load_to_lds, TENSORcnt) issued by wave 0.
// Phase 2: cross-tile union-find merge (boundary pairs only, atomicMin hook).
// Phase 3: flatten (path compression).
// Phase 4: ordered root selection (async global->LDS chunk staging + LDS scans
//          => matches scipy label order = ascending first-pixel index).
// Phase 5: per-component box stats via binary search + global atomics.
// Phase 6: keep-filter (drop last component, drop width/height < 0.01),
//          stable compaction, write boxes/scores/n_keep.
// ---------------------------------------------------------------------------

#define HH    512
#define WW    512
#define NPIX  (HH * WW)          // 262144
#define NIMG  16
#define MAXB  4096
#define BIGV  NPIX               // background sentinel (matches reference BIG)
#define TILE  32
#define TPR   (WW / TILE)        // tiles per row = 16
#define TPI   ((HH / TILE) * (WW / TILE)) // tiles per image = 256

typedef uint32_t u32x4 __attribute__((ext_vector_type(4)));
typedef uint32_t u32x8 __attribute__((ext_vector_type(8)));

// ---------------- union-find helpers (global memory) -----------------------

__device__ __forceinline__ int find_root(const int* __restrict__ P, int x) {
  int p = P[x];
  while (p != x) { x = p; p = P[x]; }
  return x;
}

__device__ __forceinline__ void merge_roots(int* P, int a, int b) {
  a = find_root(P, a);
  b = find_root(P, b);
  while (a != b) {
    if (a < b) { int t = a; a = b; b = t; }   // ensure a > b
    int old = atomicMin(&P[a], b);            // hook larger root under smaller
    if (old == a) break;                      // we were the one to hook it
    a = old;                                  // someone hooked first: keep merging
  }
}

// ---------------- Phase 1: tile-local CCL in LDS ----------------------------

__global__ void k_tile_ccl(const float* __restrict__ mask, int* __restrict__ labels) {
  __shared__ float s_mask[TILE][TILE];
  __shared__ int   s_lab[TILE + 2][TILE + 2];   // BIG halo -> no bounds checks
  __shared__ int   s_changed;

  const int tx = threadIdx.x, ty = threadIdx.y;
  const int tid = ty * TILE + tx;
  const int img  = blockIdx.x / TPI;
  const int tile = blockIdx.x % TPI;
  const int r0 = (tile / TPR) * TILE;
  const int c0 = (tile % TPR) * TILE;
  const int r = r0 + ty, c = c0 + tx;
  const int gidx = r * WW + c;

  // --- CDNA5 Tensor Data Mover: one DMA moves the whole 32x32 fp32 tile
  // global->LDS. D# group0/group1 built in SGPRs (all wave-uniform); wave 0
  // issues; TDM ignores EXEC so exactly one issue per tile. Overlapped with
  // the LDS halo init below, then published via the workgroup barrier.
  if (ty == 0) {
    const uint64_t ga =
        (uint64_t)(uintptr_t)(mask + (size_t)img * NPIX + (size_t)r0 * WW + c0);
    const uint32_t lbase = (uint32_t)(uintptr_t)&s_mask[0][0];
    u32x4 g0;
    g0[0] = 1u;                                      // count=1, user mode
    g0[1] = lbase;                                   // lds_addr
    g0[2] = (uint32_t)ga;                            // global_addr[31:0]
    g0[3] = (uint32_t)((ga >> 32) & 0x01FFFFFFu)     // global_addr[56:32]
            | (2u << 30);                            // type = 2 ("image")
    u32x8 g1;
    g1[0] = 2u << 16;                                // data_size = 4 bytes
    g1[1] = ((uint32_t)WW & 0xFFFFu) << 16;          // tensor_dim0 = 512 (lo16)
    g1[2] = ((uint32_t)HH & 0xFFFFu) << 16;          // tensor_dim1 = 512 (lo16)
    g1[3] = (uint32_t)TILE << 16;                    // tile_dim0 = 32
    g1[4] = (uint32_t)TILE;                          // tile_dim1 = 32, tile_dim2=0
    g1[5] = (uint32_t)WW;                            // tensor_dim0_stride = 512
    g1[6] = ((uint32_t)NPIX & 0xFFFFu) << 16;        // dim1_stride[15:0] (=0)
    g1[7] = (uint32_t)NPIX >> 16;                    // dim1_stride[47:16]
    asm volatile("tensor_load_to_lds %0, %1" :: "s"(g0), "s"(g1) : "memory");
  }

  // init halo + body to BIG while the DMA streams in
  for (int i = tid; i < (TILE + 2) * (TILE + 2); i += TILE * TILE)
    ((int*)s_lab)[i] = BIGV;

  if (ty == 0) __builtin_amdgcn_s_wait_tensorcnt(0); // wave-uniform branch
  __syncthreads();

  const bool fg = s_mask[ty][tx] > 0.5f;
  s_lab[ty + 1][tx + 1] = fg ? gidx : BIGV;
  __syncthreads();

  // min-neighbor propagation + pointer jumping inside the tile until stable
  for (;;) {
    if (tid == 0) s_changed = 0;
    __syncthreads();
    int l = s_lab[ty + 1][tx + 1];
    if (l != BIGV) {
      int nb = min(min(s_lab[ty][tx + 1], s_lab[ty + 2][tx + 1]),
                   min(s_lab[ty + 1][tx], s_lab[ty + 1][tx + 2]));
      int nl = min(l, nb);
      // labels always reference in-tile foreground pixels -> jump one hop
      int lr = nl / WW - r0, lc = nl % WW - c0;
      nl = min(nl, s_lab[lr + 1][lc + 1]);
      if (nl < l) { s_lab[ty + 1][tx + 1] = nl; s_changed = 1; }
    }
    __syncthreads();
    if (!s_changed) break;
  }

  labels[(size_t)img * NPIX + gidx] = s_lab[ty + 1][tx + 1];
}

// ---------------- Phase 2: cross-tile boundary merge ------------------------
// Only pixel pairs straddling a tile edge need global merging:
// 15 boundary columns x 512 rows + 15 boundary rows x 512 cols per image.

#define NBND   ((TPR - 1) * HH)          // 7680 pairs per direction per image
#define PERIMG (2 * NBND)                // 15360

__global__ void k_merge(int* __restrict__ labels) {
  int t = blockIdx.x * blockDim.x + threadIdx.x;
  if (t >= NIMG * PERIMG) return;
  const int img = t / PERIMG;
  const int u   = t % PERIMG;
  int* L = labels + (size_t)img * NPIX;
  if (u < NBND) {                         // horizontal pair across vertical edge
    const int c = (TILE - 1) + TILE * (u / HH);
    const int r = u % HH;
    const int p = r * WW + c;
    if (L[p] != BIGV && L[p + 1] != BIGV) merge_roots(L, p, p + 1);
  } else {                                // vertical pair across horizontal edge
    const int v = u - NBND;
    const int r = (TILE - 1) + TILE * (v / WW);
    const int c = v % WW;
    const int p = r * WW + c;
    __builtin_prefetch(&L[p + WW], 0, 0); // global_prefetch_b8
    if (L[p] != BIGV && L[p + WW] != BIGV) merge_roots(L, p, p + WW);
  }
}

// ---------------- Phase 3: flatten to roots ---------------------------------

__global__ void k_flatten(int* __restrict__ labels) {
  long long t = (long long)blockIdx.x * blockDim.x + threadIdx.x;
  if (t >= (long long)NIMG * NPIX) return;
  const int img = (int)(t / NPIX), p = (int)(t % NPIX);
  int* L = labels + (size_t)img * NPIX;
  int par = L[p];
  if (par == BIGV) return;
  __builtin_prefetch(&L[par], 0, 0);           // prefetch the chain head
  L[p] = find_root(L, p);
}

// ---------------- Phase 4: ordered root selection ---------------------------
// One block per image. Chunks of 1024 labels are staged to LDS with async
// global->LDS loads (ASYNCcnt); each thread loads and reads only its own slot
// so no extra barrier is needed around the staging. Deterministic prefix scan
// in raster order => root ordinals match scipy label order.

__global__ void k_select_roots(const int* __restrict__ labels,
                               int* __restrict__ sel, int* __restrict__ ncomp) {
  const int img = blockIdx.x;
  const int* L = labels + (size_t)img * NPIX;
  __shared__ int s_vals[1024];
  __shared__ int s_scan[1024];
  __shared__ int s_base;
  if (threadIdx.x == 0) s_base = 0;
  __syncthreads();
  for (int chunk = 0; chunk < NPIX; chunk += 1024) {
    const int i = chunk + (int)threadIdx.x;
    {
      unsigned long long ga = (unsigned long long)(uintptr_t)(L + i);
      unsigned int       la = (unsigned int)(uintptr_t)&s_vals[threadIdx.x];
      asm volatile("global_load_async_to_lds_b32 %0, %1, off"
                   :: "v"(la), "v"(ga) : "memory");
      asm volatile("s_wait_asynccnt 0x0" ::: "memory");
    }
    const int isRoot = (s_vals[threadIdx.x] == i) ? 1 : 0; // bg: L==BIGV != i
    s_scan[threadIdx.x] = isRoot;
    __syncthreads();
    for (int off = 1; off < 1024; off <<= 1) { // Hillis-Steele inclusive scan
      int v = s_scan[threadIdx.x];
      int a = (threadIdx.x >= (unsigned)off) ? s_scan[threadIdx.x - off] : 0;
      __syncthreads();
      s_scan[threadIdx.x] = v + a;
      __syncthreads();
    }
    const int incl = s_scan[threadIdx.x];
    const int rank = s_base + incl - isRoot;
    if (isRoot && rank < MAXB) sel[img * MAXB + rank] = i;
    __syncthreads();
    if (threadIdx.x == 1023) s_base += incl;
    __syncthreads();
  }
  if (threadIdx.x == 0) ncomp[img] = s_base;
}

// ---------------- init compact per-component stats --------------------------

__global__ void k_init_stats(int* cnt, int* rmn, int* rmx, int* cmn, int* cmx) {
  int t = blockIdx.x * blockDim.x + threadIdx.x;
  if (t >= NIMG * MAXB) return;
  cnt[t] = 0;
  rmn[t] = 1 << 30; rmx[t] = -1;
  cmn[t] = 1 << 30; cmx[t] = -1;
}

// ---------------- Phase 5: per-component box stats --------------------------

__global__ void k_stats(const int* __restrict__ labels, const int* __restrict__ sel,
                        const int* __restrict__ ncomp,
                        int* cnt, int* rmn, int* rmx, int* cmn, int* cmx) {
  long long t = (long long)blockIdx.x * blockDim.x + threadIdx.x;
  if (t >= (long long)NIMG * NPIX) return;
  const int img = (int)(t / NPIX), p = (int)(t % NPIX);
  const int* L = labels + (size_t)img * NPIX;
  const int root = L[p];
  if (root == BIGV) return;
  const int m = min(ncomp[img], MAXB);
  const int* S = sel + img * MAXB;
  int lo = 0, hi = m - 1, k = -1;
  while (lo <= hi) {                            // sel is sorted ascending
    int mid = (lo + hi) >> 1;
    int v = S[mid];
    if (v == root) { k = mid; break; }
    if (v < root) lo = mid + 1; else hi = mid - 1;
  }
  if (k < 0) return;                            // component beyond MAXB cap
  const int r = p / WW, c = p % WW;
  const int o = img * MAXB + k;
  atomicAdd(&cnt[o], 1);
  atomicMin(&rmn[o], r); atomicMax(&rmx[o], r);
  atomicMin(&cmn[o], c); atomicMax(&cmx[o], c);
}

// ---------------- Phase 6: keep-filter + stable compaction ------------------

__global__ void k_finalize(const int* __restrict__ ncomp,
                           const int* __restrict__ cnt,
                           const int* __restrict__ rmn, const int* __restrict__ rmx,
                           const int* __restrict__ cmn, const int* __restrict__ cmx,
                           float* __restrict__ out) {
  const int img = blockIdx.x;
  float* boxes  = out + (size_t)img * MAXB * 4;
  float* scores = out + (size_t)NIMG * MAXB * 4 + (size_t)img * MAXB;
  float* nk     = out + (size_t)NIMG * MAXB * 4 + (size_t)NIMG * MAXB + img;
  const int n = ncomp[img];
  const int m = min(n, MAXB);

  // zero-fill everything first (d_out is poisoned by the harness)
  for (int j = threadIdx.x; j < MAXB; j += blockDim.x) {
    scores[j] = 0.0f;
    boxes[4 * j + 0] = 0.0f; boxes[4 * j + 1] = 0.0f;
    boxes[4 * j + 2] = 0.0f; boxes[4 * j + 3] = 0.0f;
  }
  __syncthreads();

  __shared__ int s_scan[1024];
  __shared__ int s_base;
  if (threadIdx.x == 0) s_base = 0;
  __syncthreads();

  for (int chunk = 0; chunk < MAXB; chunk += 1024) {
    const int j = chunk + (int)threadIdx.x;
    int keep = 0, x1 = 0, y1 = 0, x2 = 0, y2 = 0;
    float sc = 0.0f;
    if (j < m) {
      const int o = img * MAXB + j;
      x1 = cmn[o]; x2 = cmx[o]; y1 = rmn[o]; y2 = rmx[o];
      const float cf = (float)cnt[o];
      sc = (cf / cf - 0.5f) / 0.5f;             // vsum == cnt (binary mask) => 1.0
      keep = ((float)(x2 - x1) >= 0.01f) &&     // remove_small_boxes
             ((float)(y2 - y1) >= 0.01f) &&
             (j < n - 1);                       // reference drops last component
    }
    s_scan[threadIdx.x] = keep;
    __syncthreads();
    for (int off = 1; off < 1024; off <<= 1) {
      int v = s_scan[threadIdx.x];
      int a = (threadIdx.x >= (unsigned)off) ? s_scan[threadIdx.x - off] : 0;
      __syncthreads();
      s_scan[threadIdx.x] = v + a;
      __syncthreads();
    }
    const int incl = s_scan[threadIdx.x];
    const int pos = s_base + incl - keep;
    if (keep) {
      boxes[4 * pos + 0] = (float)x1;
      boxes[4 * pos + 1] = (float)y1;
      boxes[4 * pos + 2] = (float)x2;
      boxes[4 * pos + 3] = (float)y2;
      scores[pos] = sc;
    }
    __syncthreads();
    if (threadIdx.x == 1023) s_base += incl;
    __syncthreads();
  }
  if (threadIdx.x == 0) *nk = (float)s_base;
}

// ---------------------------------------------------------------------------

extern "C" void kernel_launch(void* const* d_in, const int* in_sizes, int n_in,
                              void* d_out, int out_size, void* d_ws, size_t ws_size,
                              hipStream_t stream) {
  const float* mask = (const float*)d_in[0];   // [16,1,512,512] fp32
  float* out = (float*)d_out;                  // boxes | scores | n_keep

  // workspace carve-up (~18.3 MB)
  int* labels = (int*)d_ws;                    // NIMG*NPIX
  int* sel    = labels + (size_t)NIMG * NPIX;  // NIMG*MAXB
  int* ncomp  = sel    + (size_t)NIMG * MAXB;  // NIMG
  int* cnt    = ncomp  + NIMG;                 // NIMG*MAXB
  int* rmn    = cnt    + (size_t)NIMG * MAXB;
  int* rmx    = rmn    + (size_t)NIMG * MAXB;
  int* cmn    = rmx    + (size_t)NIMG * MAXB;
  int* cmx    = cmn    + (size_t)NIMG * MAXB;

  const int tot = NIMG * NPIX;                 // 4,194,304 pixels

  dim3 tileBlock(TILE, TILE);                  // 1024 thr = 32 waves of wave32
  k_tile_ccl   <<<NIMG * TPI, tileBlock, 0, stream>>>(mask, labels);
  k_merge      <<<(NIMG * PERIMG + 1023) / 1024, 1024, 0, stream>>>(labels);
  k_flatten    <<<tot / 1024, 1024, 0, stream>>>(labels);
  k_init_stats <<<(NIMG * MAXB + 1023) / 1024, 1024, 0, stream>>>(cnt, rmn, rmx, cmn, cmx);
  k_select_roots<<<NIMG, 1024, 0, stream>>>(labels, sel, ncomp);
  k_stats      <<<tot / 1024, 1024, 0, stream>>>(labels, sel, ncomp, cnt, rmn, rmx, cmn, cmx);
  k_finalize   <<<NIMG, 1024, 0, stream>>>(ncomp, cnt, rmn, rmx, cmn, cmx, out);
}